// SAM_21844203668081
// MI455X (gfx1250) — compile-verified
//
#include <hip/hip_runtime.h>

#define CIN_  256
#define H_    56
#define W_    56
#define HW_   (H_ * W_)
#define NB_   4
#define REL_  32
#define OUTC_ 256
#define KW_   7
#define K2_   49
#define PADR_ 3
#define EPSB_ 1e-5f

typedef float v2f __attribute__((ext_vector_type(2)));
typedef float v8f __attribute__((ext_vector_type(8)));

__device__ __forceinline__ v8f wmma_f32_16x16x4(v2f a, v2f b, v8f c) {
  // 8 args: (neg_a, A, neg_b, B, c_mod, C, reuse_a, reuse_b)
  return __builtin_amdgcn_wmma_f32_16x16x4_f32(false, a, false, b, (short)0, c,
                                               false, false);
}

__device__ __forceinline__ int reflect_idx(int q, int n) {
  if (q < 0) q = -q;
  if (q >= n) q = 2 * (n - 1) - q;
  return q;
}

// ---------------------------------------------------------------------------
// Phase A: x123[n][0:320][hw] = [w1;w2;w3](320x256) * x[n](256x3136) + bias
// One wave computes one 16x16 C tile via V_WMMA_F32_16X16X4_F32, K = 256.
// ---------------------------------------------------------------------------
__global__ __launch_bounds__(256) void sam_gemm_x123(
    const float* __restrict__ x,
    const float* __restrict__ w1, const float* __restrict__ b1,
    const float* __restrict__ w2, const float* __restrict__ b2,
    const float* __restrict__ w3, const float* __restrict__ b3,
    float* __restrict__ x123) {
  const int lane = threadIdx.x & 31;
  const int wave = blockIdx.x * (blockDim.x >> 5) + (threadIdx.x >> 5);
  const int MT = 320 / 16;   // 20
  const int NT = HW_ / 16;   // 196
  if (wave >= NB_ * MT * NT) return;  // wave-uniform; grid is exact anyway

  int t = wave;
  const int n = t / (MT * NT); t -= n * (MT * NT);
  const int mt = t / NT;
  const int nt = t - mt * NT;

  const int mrow  = lane & 15;
  const int khalf = (lane < 16) ? 0 : 2;   // A/B K sub-row per lane half
  const int row   = mt * 16 + mrow;        // 0..319
  const int col   = nt * 16 + mrow;        // 0..3135

  // Each 16-row tile lies entirely inside one weight segment (boundaries 32, 64)
  const float* arow;
  if (row < REL_)           arow = w1 + row * CIN_;
  else if (row < 2 * REL_)  arow = w2 + (row - REL_) * CIN_;
  else                      arow = w3 + (row - 2 * REL_) * CIN_;

  const float* bcol = x + (size_t)n * CIN_ * HW_ + col;

  v8f acc = {0.f, 0.f, 0.f, 0.f, 0.f, 0.f, 0.f, 0.f};
  for (int k = 0; k < CIN_; k += 4) {
    v2f a, b;
    a.x = arow[k + khalf];
    a.y = arow[k + khalf + 1];
    b.x = bcol[(size_t)(k + khalf) * HW_];
    b.y = bcol[(size_t)(k + khalf + 1) * HW_];
    acc = wmma_f32_16x16x4(a, b, acc);
  }

  // C/D layout: vgpr i -> M = i (lanes 0-15) / i+8 (lanes 16-31), N = lane%16
  const int rbase = mt * 16 + ((lane < 16) ? 0 : 8);
  float* obase = x123 + (size_t)n * 320 * HW_ + (size_t)(nt * 16 + mrow);
  for (int i = 0; i < 8; ++i) {
    int r = rbase + i;
    float bias;
    if (r < REL_)           bias = b1[r];
    else if (r < 2 * REL_)  bias = b2[r - REL_];
    else                    bias = b3[r - 2 * REL_];
    obase[(size_t)r * HW_] = acc[i] + bias;
  }
}

// ---------------------------------------------------------------------------
// Phase B: one wave = one pixel. Fully fused: feat -> BN1/ReLU -> cw1 GEMM
// -> BN2/ReLU -> cw2 GEMM (+bias) -> softmax(49) -> aggregate x3 patches.
// GEMMs run as 2x4 tiles of 16x16 WMMA (K padded 34->36 / 32, N padded 49->64).
// ---------------------------------------------------------------------------
__global__ __launch_bounds__(32) void sam_attn(
    const float* __restrict__ x123, const float* __restrict__ wp,
    const float* __restrict__ bn1_g, const float* __restrict__ bn1_b,
    const float* __restrict__ bn1_m, const float* __restrict__ bn1_v,
    const float* __restrict__ cw1,
    const float* __restrict__ bn2_g, const float* __restrict__ bn2_b,
    const float* __restrict__ bn2_m, const float* __restrict__ bn2_v,
    const float* __restrict__ cw2, const float* __restrict__ cb2,
    float* __restrict__ out) {
  __shared__ float sFeat[36][64];   // BN1+ReLU'd feature block (zero padded)
  __shared__ float sW1[32][36];     // cw1 zero-padded K 34->36
  __shared__ float sW2[32][32];     // cw2
  __shared__ float sY[32][64];      // BN2+ReLU'd mid activation
  __shared__ float sWgt[32][64];    // logits -> softmax weights
  __shared__ int   sNbr[K2_];       // reflected neighbor pixel index
  __shared__ float sX1c[REL_];
  __shared__ float sBn1s[REL_ + 2], sBn1o[REL_ + 2];
  __shared__ float sBn2s[REL_],     sBn2o[REL_];

  const int tid = threadIdx.x;
  const int pix = blockIdx.x;
  const int n  = pix / HW_;
  const int hw = pix - n * HW_;
  const int h  = hw / W_;
  const int w  = hw - h * W_;

  const float* xb = x123 + (size_t)n * 320 * HW_;

  // ---- setup: zero padding, folded BN params, weights, center x1 ----
  for (int i = tid; i < 36 * 64; i += 32) (&sFeat[0][0])[i] = 0.f;
  for (int i = tid; i < REL_ + 2; i += 32) {
    float s = bn1_g[i] * rsqrtf(bn1_v[i] + EPSB_);
    sBn1s[i] = s;
    sBn1o[i] = bn1_b[i] - bn1_m[i] * s;
  }
  if (tid < REL_) {
    float s = bn2_g[tid] * rsqrtf(bn2_v[tid] + EPSB_);
    sBn2s[tid] = s;
    sBn2o[tid] = bn2_b[tid] - bn2_m[tid] * s;
    sX1c[tid] = xb[(size_t)tid * HW_ + hw];
  }
  for (int i = tid; i < REL_ * 36; i += 32) {
    int c = i / 36, ch = i - c * 36;
    sW1[c][ch] = (ch < REL_ + 2) ? cw1[c * (REL_ + 2) + ch] : 0.f;
  }
  for (int i = tid; i < REL_ * REL_; i += 32) (&sW2[0][0])[i] = cw2[i];
  __syncthreads();

  // ---- neighbors + analytic position-difference channels (bias cancels) ----
  const float wp00 = wp[0], wp01 = wp[1], wp10 = wp[2], wp11 = wp[3];
  for (int k = tid; k < K2_; k += 32) {
    int ki = k / KW_;
    int di = ki - PADR_, dj = (k - ki * KW_) - PADR_;
    int hr = reflect_idx(h + di, H_);
    int wr = reflect_idx(w + dj, W_);
    sNbr[k] = hr * W_ + wr;
    float dlw = 2.f * (float)(w - wr) / (float)(W_ - 1);
    float dlh = 2.f * (float)(h - hr) / (float)(H_ - 1);
    float p0 = wp00 * dlw + wp01 * dlh;
    float p1 = wp10 * dlw + wp11 * dlh;
    sFeat[REL_ + 0][k] = fmaxf(p0 * sBn1s[REL_ + 0] + sBn1o[REL_ + 0], 0.f);
    sFeat[REL_ + 1][k] = fmaxf(p1 * sBn1s[REL_ + 1] + sBn1o[REL_ + 1], 0.f);
  }
  __syncthreads();

  // ---- Subtraction2 channels 0..31, BN1 + ReLU fused ----
  const float* x2b = xb + (size_t)REL_ * HW_;
  for (int i = tid; i < REL_ * K2_; i += 32) {
    int c = i / K2_, k = i - c * K2_;
    float v = sX1c[c] - x2b[(size_t)c * HW_ + sNbr[k]];
    sFeat[c][k] = fmaxf(v * sBn1s[c] + sBn1o[c], 0.f);
  }
  __syncthreads();

  const int mrow  = tid & 15;
  const int khalf = (tid < 16) ? 0 : 2;
  const int rhalf = (tid < 16) ? 0 : 8;

  // ---- GEMM1: sY = relu(bn2(cw1 (32x36) * sFeat (36x64))) ----
  for (int mt = 0; mt < 2; ++mt) {
    for (int nt = 0; nt < 4; ++nt) {
      v8f acc = {0.f, 0.f, 0.f, 0.f, 0.f, 0.f, 0.f, 0.f};
      for (int kk = 0; kk < 36; kk += 4) {
        v2f a, b;
        a.x = sW1[mt * 16 + mrow][kk + khalf];
        a.y = sW1[mt * 16 + mrow][kk + khalf + 1];
        b.x = sFeat[kk + khalf][nt * 16 + mrow];
        b.y = sFeat[kk + khalf + 1][nt * 16 + mrow];
        acc = wmma_f32_16x16x4(a, b, acc);
      }
      int col = nt * 16 + mrow;
      for (int i = 0; i < 8; ++i) {
        int r = mt * 16 + rhalf + i;
        sY[r][col] = fmaxf(acc[i] * sBn2s[r] + sBn2o[r], 0.f);
      }
    }
  }
  __syncthreads();

  // ---- GEMM2: sWgt = cw2 (32x32) * sY (32x64) + cb2 ----
  for (int mt = 0; mt < 2; ++mt) {
    for (int nt = 0; nt < 4; ++nt) {
      v8f acc = {0.f, 0.f, 0.f, 0.f, 0.f, 0.f, 0.f, 0.f};
      for (int kk = 0; kk < 32; kk += 4) {
        v2f a, b;
        a.x = sW2[mt * 16 + mrow][kk + khalf];
        a.y = sW2[mt * 16 + mrow][kk + khalf + 1];
        b.x = sY[kk + khalf][nt * 16 + mrow];
        b.y = sY[kk + khalf + 1][nt * 16 + mrow];
        acc = wmma_f32_16x16x4(a, b, acc);
      }
      int col = nt * 16 + mrow;
      for (int i = 0; i < 8; ++i) {
        int r = mt * 16 + rhalf + i;
        sWgt[r][col] = acc[i] + cb2[r];
      }
    }
  }
  __syncthreads();

  // ---- softmax over the 49 real window slots; one lane per loc channel ----
  {
    float mx = -3.402823466e38f;
    for (int k = 0; k < K2_; ++k) mx = fmaxf(mx, sWgt[tid][k]);
    float sum = 0.f;
    for (int k = 0; k < K2_; ++k) {
      float e = __expf(sWgt[tid][k] - mx);
      sWgt[tid][k] = e;
      sum += e;
    }
    float inv = 1.f / sum;
    for (int k = 0; k < K2_; ++k) sWgt[tid][k] *= inv;
  }
  __syncthreads();

  // ---- aggregation: out[c] = sum_k x3[c, nbr_k] * wgt[c>>3, k] ----
  const float* x3b = xb + (size_t)(2 * REL_) * HW_;
  float* ob = out + (size_t)n * OUTC_ * HW_ + hw;
  for (int c = tid; c < OUTC_; c += 32) {
    const float* xr = x3b + (size_t)c * HW_;
    const float* wr = &sWgt[c >> 3][0];
    float acc = 0.f;
    for (int k = 0; k < K2_; ++k) acc += xr[sNbr[k]] * wr[k];
    ob[(size_t)c * HW_] = acc;
  }
}

extern "C" void kernel_launch(void* const* d_in, const int* in_sizes, int n_in,
                              void* d_out, int out_size, void* d_ws, size_t ws_size,
                              hipStream_t stream) {
  (void)in_sizes; (void)n_in; (void)out_size; (void)ws_size;
  const float* x     = (const float*)d_in[0];
  const float* w1    = (const float*)d_in[1];
  const float* b1    = (const float*)d_in[2];
  const float* w2    = (const float*)d_in[3];
  const float* b2    = (const float*)d_in[4];
  const float* w3    = (const float*)d_in[5];
  const float* b3    = (const float*)d_in[6];
  const float* wp    = (const float*)d_in[7];
  /* d_in[8] = bp: cancels in the position subtraction, unused */
  const float* bn1_g = (const float*)d_in[9];
  const float* bn1_b = (const float*)d_in[10];
  const float* bn1_m = (const float*)d_in[11];
  const float* bn1_v = (const float*)d_in[12];
  const float* cw1   = (const float*)d_in[13];
  const float* bn2_g = (const float*)d_in[14];
  const float* bn2_b = (const float*)d_in[15];
  const float* bn2_m = (const float*)d_in[16];
  const float* bn2_v = (const float*)d_in[17];
  const float* cw2   = (const float*)d_in[18];
  const float* cb2   = (const float*)d_in[19];

  float* x123 = (float*)d_ws;  // [4][320][3136] fp32 = 16.06 MB (L2-resident)

  const int tiles   = NB_ * (320 / 16) * (HW_ / 16);  // 15680 waves
  const int blocksA = tiles / 8;                      // 8 waves per 256-thr block
  sam_gemm_x123<<<blocksA, 256, 0, stream>>>(x, w1, b1, w2, b2, w3, b3, x123);
  sam_attn<<<NB_ * HW_, 32, 0, stream>>>(x123, wp, bn1_g, bn1_b, bn1_m, bn1_v,
                                         cw1, bn2_g, bn2_b, bn2_m, bn2_v,
                                         cw2, cb2, (float*)d_out);
}